// UPSSITS_28312424415434
// MI455X (gfx1250) — compile-verified
//
#include <hip/hip_runtime.h>

typedef float v2f __attribute__((ext_vector_type(2)));
typedef float v8f __attribute__((ext_vector_type(8)));

#define N_CLASSES 20
#define FEAT 4097
#define NHID 128
#define BT 24          // B*T = 2*12
#define HW 16384       // 128*128
#define NC 10          // channels
#define TT 12
#define KSLICES 8      // K-split of GEMM1 (4096 = 8 * 512)
#define KSPAN 512      // f-range per slice

__device__ __forceinline__ int imin(int a, int b) { return a < b ? a : b; }

// ---------------------------------------------------------------------------
// Kernel 1: build feat (BT x 4097) from att (16,B,T,16,16) + batch_positions
// ---------------------------------------------------------------------------
__global__ void feat_kernel(const float* __restrict__ att,
                            const int* __restrict__ pos,
                            float* __restrict__ feat) {
    long i = (long)blockIdx.x * 256 + threadIdx.x;
    if (i >= (long)BT * FEAT) return;
    int m = (int)(i / FEAT);
    int f = (int)(i % FEAT);
    int b = m / TT, t = m % TT;
    float v;
    if (f < 4096) {
        int head = f >> 8;          // 16 heads
        int r    = f & 255;         // i*16+j
        v = att[(((long)head * 2 + b) * TT + t) * 256 + r];
    } else {
        v = (float)pos[b * TT + t];
    }
    feat[i] = v;
}

// ---------------------------------------------------------------------------
// Kernel 2: GEMM1 partials. Grid = 20*8 blocks (kcls*8 + kslice),
// block = 256 (8 waves = 8 N-tiles). Each wave holds BOTH M-tiles (2
// accumulators) so every W1 load is used twice -> W1 (42 MB) streamed once.
// K split 8 ways for memory-level parallelism (1280 waves total).
// fp32 WMMA 16x16x4; f = 4096 tail handled in the reduction kernel.
// ---------------------------------------------------------------------------
__global__ __launch_bounds__(256) void gemm1_kernel(const float* __restrict__ feat,
                                                    const float* __restrict__ W1,
                                                    float* __restrict__ P) {
    int blk   = blockIdx.x;
    int kcls  = blk >> 3;
    int slice = blk & 7;
    int wave  = threadIdx.x >> 5;   // N-tile 0..7
    int lane  = threadIdx.x & 31;
    int n0    = wave * 16;
    int l15   = lane & 15;
    int hi    = lane >> 4;          // which K half-pair this lane holds
    int col   = n0 + l15;

    const float* fA0 = feat + (long)l15 * FEAT;                 // rows 0..15
    const float* fA1 = feat + (long)imin(16 + l15, BT - 1) * FEAT; // rows 16..23 (clamped)
    const float* fB  = W1 + (long)kcls * FEAT * NHID + col;

    int f0 = slice * KSPAN;
    v8f acc0 = {};
    v8f acc1 = {};
    for (int s = 0; s < KSPAN / 4; ++s) {
        int kk = f0 + 4 * s + 2 * hi;
        v2f b;  b.x  = fB[(long)kk * NHID]; b.y  = fB[(long)(kk + 1) * NHID];
        v2f a0; a0.x = fA0[kk];             a0.y = fA0[kk + 1];
        v2f a1; a1.x = fA1[kk];             a1.y = fA1[kk + 1];
        acc0 = __builtin_amdgcn_wmma_f32_16x16x4_f32(false, a0, false, b,
                                                     (short)0, acc0, false, false);
        acc1 = __builtin_amdgcn_wmma_f32_16x16x4_f32(false, a1, false, b,
                                                     (short)0, acc1, false, false);
    }
    // partials: P[((kcls*8+slice)*BT + row)*NHID + col]
    float* Pp = P + ((long)(kcls * KSLICES + slice) * BT) * NHID + col;
#pragma unroll
    for (int r = 0; r < 8; ++r) {
        int row0 = r + 8 * hi;          // 0..15, always valid
        int row1 = 16 + r + 8 * hi;     // 16..31, valid < 24
        Pp[(long)row0 * NHID] = acc0[r];
        if (row1 < BT) Pp[(long)row1 * NHID] = acc1[r];
    }
}

// ---------------------------------------------------------------------------
// Kernel 3: reduce GEMM1 partials + rank-1 tail (f=4096) + bias + ReLU -> H1
// ---------------------------------------------------------------------------
__global__ void reduce1_kernel(const float* __restrict__ P,
                               const float* __restrict__ feat,
                               const float* __restrict__ W1,
                               const float* __restrict__ b1,
                               float* __restrict__ H1) {
    int idx = blockIdx.x * 256 + threadIdx.x;       // k*BT*NHID total
    if (idx >= N_CLASSES * BT * NHID) return;
    int col = idx & (NHID - 1);
    int row = (idx >> 7) % BT;
    int k   = idx / (BT * NHID);
    float s = 0.f;
#pragma unroll
    for (int sl = 0; sl < KSLICES; ++sl)
        s += P[((long)(k * KSLICES + sl) * BT + row) * NHID + col];
    s = fmaf(feat[(long)row * FEAT + 4096],
             W1[((long)k * FEAT + 4096) * NHID + col], s);
    s += b1[k * NHID + col];
    H1[idx] = s > 0.f ? s : 0.f;
}

// ---------------------------------------------------------------------------
// Kernel 4: H2[k] = relu(H1[k](24x128) @ W2[k](128x128) + b2[k])
// ---------------------------------------------------------------------------
__global__ __launch_bounds__(256) void gemm2_kernel(const float* __restrict__ H1,
                                                    const float* __restrict__ W2,
                                                    const float* __restrict__ b2,
                                                    float* __restrict__ H2) {
    int blk  = blockIdx.x;
    int kcls = blk >> 1;
    int m0   = (blk & 1) * 16;
    int wave = threadIdx.x >> 5;
    int lane = threadIdx.x & 31;
    int n0   = wave * 16;
    int l15  = lane & 15;
    int hi   = lane >> 4;
    int rowA = imin(m0 + l15, BT - 1);
    int col  = n0 + l15;

    const float* fA = H1 + ((long)kcls * BT + rowA) * NHID;
    const float* fB = W2 + (long)kcls * NHID * NHID + col;

    v8f acc = {};
    for (int s = 0; s < 32; ++s) {
        int kk = 4 * s + 2 * hi;
        v2f a; a.x = fA[kk];              a.y = fA[kk + 1];
        v2f b; b.x = fB[(long)kk * NHID]; b.y = fB[(long)(kk + 1) * NHID];
        acc = __builtin_amdgcn_wmma_f32_16x16x4_f32(false, a, false, b,
                                                    (short)0, acc, false, false);
    }
    float bias = b2[kcls * NHID + col];
#pragma unroll
    for (int r = 0; r < 8; ++r) {
        int row = m0 + r + 8 * hi;
        float v = acc[r] + bias;
        v = v > 0.0f ? v : 0.0f;
        if (row < BT) H2[((long)kcls * BT + row) * NHID + col] = v;
    }
}

// ---------------------------------------------------------------------------
// Kernel 5: beta[k] = H2[k](24x128) @ W3[k](128x20) + b3[k]
// grid = 40 (kcls*2+mtile), block = 64 (2 waves = 2 N-tiles; cols >= 20 clamped)
// ---------------------------------------------------------------------------
__global__ __launch_bounds__(64) void gemm3_kernel(const float* __restrict__ H2,
                                                   const float* __restrict__ W3,
                                                   const float* __restrict__ b3,
                                                   float* __restrict__ beta) {
    int blk  = blockIdx.x;
    int kcls = blk >> 1;
    int m0   = (blk & 1) * 16;
    int wave = threadIdx.x >> 5;
    int lane = threadIdx.x & 31;
    int n0   = wave * 16;
    int l15  = lane & 15;
    int hi   = lane >> 4;
    int rowA = imin(m0 + l15, BT - 1);
    int col  = n0 + l15;
    int cc   = imin(col, 19);           // clamp to stay in-bounds of W3

    const float* fA = H2 + ((long)kcls * BT + rowA) * NHID;
    const float* fB = W3 + (long)kcls * NHID * 20 + cc;

    v8f acc = {};
    for (int s = 0; s < 32; ++s) {
        int kk = 4 * s + 2 * hi;
        v2f a; a.x = fA[kk];            a.y = fA[kk + 1];
        v2f b; b.x = fB[(long)kk * 20]; b.y = fB[(long)(kk + 1) * 20];
        acc = __builtin_amdgcn_wmma_f32_16x16x4_f32(false, a, false, b,
                                                    (short)0, acc, false, false);
    }
    float bias = b3[kcls * 20 + cc];
#pragma unroll
    for (int r = 0; r < 8; ++r) {
        int row = m0 + r + 8 * hi;
        if (row < BT && col < 20)
            beta[((long)kcls * BT + row) * 20 + col] = acc[r] + bias;
    }
}

// ---------------------------------------------------------------------------
// Kernel 6: per-(b,c,k) coefficients: a=1+w, bb, and Qa=Σa², Qab=Σa·bb, Qb=Σbb²
// ---------------------------------------------------------------------------
__global__ void prep_kernel(const float* __restrict__ beta,
                            float* __restrict__ aT, float* __restrict__ bT,
                            float* __restrict__ Q) {
    int idx = blockIdx.x * 256 + threadIdx.x;      // (b*10+c)*20 + k, 400 total
    if (idx >= 400) return;
    int k  = idx % N_CLASSES;
    int bc = idx / N_CLASSES;
    int c  = bc % NC;
    int b  = bc / NC;
    float qa = 0.f, qab = 0.f, qb = 0.f;
#pragma unroll
    for (int t = 0; t < TT; ++t) {
        int m    = b * TT + t;
        float w  = beta[((long)k * BT + m) * 20 + 2 * c];
        float bb = beta[((long)k * BT + m) * 20 + 2 * c + 1];
        float a  = 1.0f + w;
        aT[(bc * N_CLASSES + k) * TT + t] = a;
        bT[(bc * N_CLASSES + k) * TT + t] = bb;
        qa  = fmaf(a, a, qa);
        qab = fmaf(a, bb, qab);
        qb  = fmaf(bb, bb, qb);
    }
    Q[(bc * N_CLASSES + k) * 3 + 0] = qa;
    Q[(bc * N_CLASSES + k) * 3 + 1] = qab;
    Q[(bc * N_CLASSES + k) * 3 + 2] = qb;
}

// ---------------------------------------------------------------------------
// Kernel 7: fused per-pixel score/argmin/reconstruct.
// score_k(p) = I·(I·Qa + 2Qab − 2Σₜaₜxₜ) + Qb − 2Σₜbbₜxₜ   (k-const terms dropped)
// grid = (64, 20): 256 pixels per block, blockIdx.y = b*10+c
// ---------------------------------------------------------------------------
__global__ __launch_bounds__(256) void pixel_kernel(const float* __restrict__ input,
                                                    const float* __restrict__ inten,
                                                    const float* __restrict__ aT,
                                                    const float* __restrict__ bT,
                                                    const float* __restrict__ Q,
                                                    float* __restrict__ out) {
    __shared__ float sa[N_CLASSES * TT];
    __shared__ float sb[N_CLASSES * TT];
    __shared__ float sq[N_CLASSES * 3];
    int bc = blockIdx.y;
    int b  = bc / NC;
    int c  = bc % NC;
    int tid = threadIdx.x;
    if (tid < N_CLASSES * TT) {
        sa[tid] = aT[(long)bc * N_CLASSES * TT + tid];
        sb[tid] = bT[(long)bc * N_CLASSES * TT + tid];
    }
    if (tid < N_CLASSES * 3) sq[tid] = Q[(long)bc * N_CLASSES * 3 + tid];
    __syncthreads();

    int p = blockIdx.x * 256 + tid;
    float I = inten[(long)b * HW + p];
    float x[TT];
#pragma unroll
    for (int t = 0; t < TT; ++t)
        x[t] = input[(((long)b * TT + t) * NC + c) * HW + p];

    float best = 3.4e38f;
    int bidx = 0;
#pragma unroll
    for (int k = 0; k < N_CLASSES; ++k) {
        float A1 = 0.f, A2 = 0.f;
#pragma unroll
        for (int t = 0; t < TT; ++t) {
            A1 = fmaf(sa[k * TT + t], x[t], A1);
            A2 = fmaf(sb[k * TT + t], x[t], A2);
        }
        float s = fmaf(I, fmaf(I, sq[k * 3 + 0], 2.f * sq[k * 3 + 1] - 2.f * A1),
                       sq[k * 3 + 2] - 2.f * A2);
        if (s < best) { best = s; bidx = k; }
    }
#pragma unroll
    for (int t = 0; t < TT; ++t)
        out[(((long)b * TT + t) * NC + c) * HW + p] =
            fmaf(sa[bidx * TT + t], I, sb[bidx * TT + t]);
}

// ---------------------------------------------------------------------------
extern "C" void kernel_launch(void* const* d_in, const int* in_sizes, int n_in,
                              void* d_out, int out_size, void* d_ws, size_t ws_size,
                              hipStream_t stream) {
    const float* input = (const float*)d_in[0];
    const int*   bpos  = (const int*)d_in[1];
    const float* inten = (const float*)d_in[2];
    const float* att   = (const float*)d_in[3];
    const float* W1    = (const float*)d_in[4];
    const float* b1    = (const float*)d_in[5];
    const float* W2    = (const float*)d_in[6];
    const float* b2    = (const float*)d_in[7];
    const float* W3    = (const float*)d_in[8];
    const float* b3    = (const float*)d_in[9];

    float* ws   = (float*)d_ws;
    float* feat = ws;                 // 24*4097 = 98328 (pad to 98336)
    float* P    = ws + 98336;         // 20*8*24*128 = 491520 (GEMM1 partials)
    float* H1   = ws + 589856;        // 20*24*128 = 61440
    float* H2   = ws + 651296;        // 61440
    float* beta = ws + 712736;        // 20*24*20 = 9600
    float* aT   = ws + 722336;        // 4800
    float* bT   = ws + 727136;        // 4800
    float* Q    = ws + 731936;        // 1200   (total ~2.93 MB)

    float* out = (float*)d_out;

    feat_kernel<<<(BT * FEAT + 255) / 256, 256, 0, stream>>>(att, bpos, feat);
    gemm1_kernel<<<N_CLASSES * KSLICES, 256, 0, stream>>>(feat, W1, P);
    reduce1_kernel<<<(N_CLASSES * BT * NHID + 255) / 256, 256, 0, stream>>>(P, feat, W1, b1, H1);
    gemm2_kernel<<<N_CLASSES * 2, 256, 0, stream>>>(H1, W2, b2, H2);
    gemm3_kernel<<<N_CLASSES * 2, 64, 0, stream>>>(H2, W3, b3, beta);
    prep_kernel<<<2, 256, 0, stream>>>(beta, aT, bT, Q);
    pixel_kernel<<<dim3(HW / 256, 2 * NC), 256, 0, stream>>>(input, inten, aT, bT, Q, out);

    // second tuple output: intensity_map passthrough
    hipMemcpyAsync(out + (long)2 * TT * NC * HW, inten,
                   (size_t)2 * HW * sizeof(float), hipMemcpyDeviceToDevice, stream);
}